// FakeNewsNet_58299886076255
// MI455X (gfx1250) — compile-verified
//
#include <hip/hip_runtime.h>
#include <hip/hip_bf16.h>

typedef __attribute__((ext_vector_type(2))) float v2f;
typedef __attribute__((ext_vector_type(8))) float v8f;

#define HIDC 64
#define INF_KEY 0x007FFFFFu   // ordered-uint key of -inf

// ---------- ordered-float <-> uint for atomic max over signed floats ----------
__device__ __forceinline__ unsigned fkey(float f) {
    unsigned u = __float_as_uint(f);
    return (u & 0x80000000u) ? ~u : (u | 0x80000000u);
}
__device__ __forceinline__ float fdec(unsigned k) {
    unsigned u = (k & 0x80000000u) ? (k & 0x7fffffffu) : ~k;
    return __uint_as_float(u);
}

// ---------- init kernels (graph-replay safe re-zeroing) ----------
__global__ void set_f32(float* __restrict__ p, float v, unsigned n) {
    unsigned i = blockIdx.x * blockDim.x + threadIdx.x;
    if (i < n) p[i] = v;
}
__global__ void set_u32(unsigned* __restrict__ p, unsigned v, unsigned n) {
    unsigned i = blockIdx.x * blockDim.x + threadIdx.x;
    if (i < n) p[i] = v;
}

// ---------- fp32 WMMA GEMM: C[M x 64] = A[M x K] * W[K x 64] ----------
// One wave per 16x16 output tile; v_wmma_f32_16x16x4_f32 along K.
// Hot loop is branch-free: row clamped once, K main loop needs no bounds
// checks (tail step handled separately with selects, not branches).
__global__ void gemm_wmma_f32(const float* __restrict__ A,
                              const float* __restrict__ W,
                              float* __restrict__ C,
                              int M, int K) {
    const int lane   = threadIdx.x & 31;
    const int wave   = blockIdx.x * (blockDim.x >> 5) + (threadIdx.x >> 5);
    const int mTiles = (M + 15) >> 4;
    if (wave >= mTiles * 4) return;          // wave-uniform exit, EXEC stays full
    const int mt = wave >> 2;                // 4 N-tiles cover the 64 cols
    const int nt = wave & 3;

    const int row   = mt * 16 + (lane & 15);     // A row for this lane
    const int rowc  = row < M ? row : (M - 1);   // clamp: loads always legal
    const int col   = nt * 16 + (lane & 15);     // B/C col for this lane
    const int khalf = (lane >> 4) * 2;           // lanes 16-31 hold K+2,K+3

    const float* __restrict__ Ap = A + (size_t)rowc * K + khalf;   // + k0
    const float* __restrict__ Bp = W + (size_t)khalf * HIDC + col; // + k0*64

    const int kmain = K & ~3;
    v8f acc = {};
#pragma unroll 2
    for (int k0 = 0; k0 < kmain; k0 += 4) {
        v2f a, b;
        a[0] = Ap[k0];                      // consecutive pair -> b64 load
        a[1] = Ap[k0 + 1];
        b[0] = Bp[(size_t)k0 * HIDC];
        b[1] = Bp[(size_t)(k0 + 1) * HIDC];
        acc = __builtin_amdgcn_wmma_f32_16x16x4_f32(
            false, a, false, b, (short)0, acc, false, false);
    }
    if (kmain < K) {                        // K % 4 tail (e.g. K=310 -> 2 cols)
        const int ka  = kmain + khalf;
        const int k0c = (ka     < K) ? ka     : (K - 1);  // clamped, always legal
        const int k1c = (ka + 1 < K) ? ka + 1 : (K - 1);
        float a0 = A[(size_t)rowc * K + k0c];
        float a1 = A[(size_t)rowc * K + k1c];
        float b0 = W[(size_t)k0c * HIDC + col];
        float b1 = W[(size_t)k1c * HIDC + col];
        v2f a, b;
        a[0] = (ka     < K) ? a0 : 0.0f;    // selects (v_cndmask), no branches
        a[1] = (ka + 1 < K) ? a1 : 0.0f;
        b[0] = (ka     < K) ? b0 : 0.0f;
        b[1] = (ka + 1 < K) ? b1 : 0.0f;
        acc = __builtin_amdgcn_wmma_f32_16x16x4_f32(
            false, a, false, b, (short)0, acc, false, false);
    }

    // C/D layout: VGPR v -> M = v + 8*(lane>>4); N = lane&15
    const int rbase = mt * 16 + 8 * (lane >> 4);
#pragma unroll
    for (int v = 0; v < 8; ++v) {
        const int r = rbase + v;
        if (r < M) C[(size_t)r * HIDC + col] = acc[v];
    }
}

// ---------- per-node attention coefficients ----------
__global__ void node_attn(const float* __restrict__ xp,
                          const float* __restrict__ asrc,
                          const float* __restrict__ adst,
                          float* __restrict__ als, float* __restrict__ ald,
                          int n) {
    int i = blockIdx.x * blockDim.x + threadIdx.x;
    if (i >= n) return;
    const float* xr = xp + (size_t)i * HIDC;
    float s = 0.0f, d = 0.0f;
#pragma unroll 8
    for (int c = 0; c < HIDC; ++c) { s += xr[c] * asrc[c]; d += xr[c] * adst[c]; }
    als[i] = s; ald[i] = d;
}

// ---------- edge pass 1: e = leaky_relu(als[s]+ald[d]); segment max ----------
__global__ void edge_max(const int* __restrict__ srcp, const int* __restrict__ dstp,
                         const float* __restrict__ als, const float* __restrict__ ald,
                         float* __restrict__ ew, unsigned* __restrict__ emax,
                         int E, int ET) {
    int i = blockIdx.x * blockDim.x + threadIdx.x;
    if (i >= ET) return;
    int s = (i < E) ? srcp[i] : (i - E);
    int d = (i < E) ? dstp[i] : (i - E);
    float e = als[s] + ald[d];
    e = (e > 0.0f) ? e : 0.2f * e;           // leaky relu, slope 0.2
    ew[i] = e;
    atomicMax(&emax[d], fkey(e));
}

// ---------- edge pass 2: w = exp(e - max); segment sum ----------
__global__ void edge_exp(const int* __restrict__ srcp, const int* __restrict__ dstp,
                         float* __restrict__ ew, const unsigned* __restrict__ emax,
                         float* __restrict__ denom, int E, int ET) {
    int i = blockIdx.x * blockDim.x + threadIdx.x;
    if (i >= ET) return;
    int d = (i < E) ? dstp[i] : (i - E);
    float w = __expf(ew[i] - fdec(emax[d]));
    ew[i] = w;
    atomicAdd(&denom[d], w);
}

// ---------- denom -> 1/(denom + eps), so the big scatter pass multiplies ----------
__global__ void recip(float* __restrict__ denom, int n) {
    int i = blockIdx.x * blockDim.x + threadIdx.x;
    if (i >= n) return;
    denom[i] = 1.0f / (denom[i] + 1e-16f);
}

// ---------- edge pass 3: out[dst] += alpha * xp[src], 64 lanes per edge ----------
__global__ void edge_aggr(const int* __restrict__ srcp, const int* __restrict__ dstp,
                          const float* __restrict__ ew, const float* __restrict__ rdenom,
                          const float* __restrict__ xp, float* __restrict__ out,
                          int E, int ET) {
    unsigned gid = blockIdx.x * blockDim.x + threadIdx.x;
    unsigned e = gid >> 6;
    if (e >= (unsigned)ET) return;
    int c = gid & 63;
    int s = (e < (unsigned)E) ? srcp[e] : (int)(e - E);
    int d = (e < (unsigned)E) ? dstp[e] : (int)(e - E);
    float alpha = ew[e] * rdenom[d];
    atomicAdd(&out[(size_t)d * HIDC + c], alpha * xp[(size_t)s * HIDC + c]);
}

// ---------- bias + relu in place ----------
__global__ void bias_relu(float* __restrict__ h, const float* __restrict__ b, unsigned n64) {
    unsigned i = blockIdx.x * blockDim.x + threadIdx.x;
    if (i >= n64) return;
    float v = h[i] + b[i & 63];
    h[i] = v > 0.0f ? v : 0.0f;
}

// ---------- global mean pool accumulation ----------
__global__ void pool_accum(const float* __restrict__ h, const int* __restrict__ batch,
                           float* __restrict__ sums, float* __restrict__ cnt, int n) {
    unsigned i = blockIdx.x * blockDim.x + threadIdx.x;
    unsigned nd = i >> 6;
    if (nd >= (unsigned)n) return;
    int c = i & 63;
    int g = batch[nd];
    atomicAdd(&sums[(size_t)g * HIDC + c], h[i]);
    if (c == 0) atomicAdd(&cnt[g], 1.0f);
}

// ---------- root = first node of each graph (batch is sorted) ----------
__global__ void root_find(const int* __restrict__ batch, int* __restrict__ root, int n) {
    int i = blockIdx.x * blockDim.x + threadIdx.x;
    if (i >= n) return;
    if (i == 0 || batch[i] != batch[i - 1]) root[batch[i]] = i;
}

// ---------- fused readout: pooled MLP + root-node MLP + concat + sigmoid ----------
__global__ void readout(const float* __restrict__ sums, const float* __restrict__ cnt,
                        const float* __restrict__ x, const int* __restrict__ root,
                        const float* __restrict__ Wr, const float* __restrict__ br,
                        const float* __restrict__ Wn, const float* __restrict__ bn,
                        const float* __restrict__ Wc, const float* __restrict__ bc,
                        float* __restrict__ out, int IN_dim) {
    __shared__ float pooled[HIDC], hr[HIDC], nw[HIDC];
    int b = blockIdx.x, c = threadIdx.x;
    float cn = cnt[b]; cn = cn > 1.0f ? cn : 1.0f;
    pooled[c] = sums[(size_t)b * HIDC + c] / cn;
    __syncthreads();

    float accr = br[c];
#pragma unroll 8
    for (int k = 0; k < HIDC; ++k) accr += pooled[k] * Wr[(size_t)c * HIDC + k];
    hr[c] = accr > 0.0f ? accr : 0.0f;

    const float* xr = x + (size_t)root[b] * IN_dim;
    float accn = bn[c];
    for (int k = 0; k < IN_dim; ++k) accn += xr[k] * Wn[(size_t)c * IN_dim + k];
    nw[c] = accn > 0.0f ? accn : 0.0f;
    __syncthreads();

    if (c == 0) {
        float z = bc[0];
#pragma unroll 8
        for (int k = 0; k < HIDC; ++k) z += hr[k] * Wc[k] + nw[k] * Wc[HIDC + k];
        out[b] = 1.0f / (1.0f + __expf(-z));
    }
}

static inline unsigned cdiv(unsigned a, unsigned b) { return (a + b - 1) / b; }

extern "C" void kernel_launch(void* const* d_in, const int* in_sizes, int n_in,
                              void* d_out, int out_size, void* d_ws, size_t ws_size,
                              hipStream_t stream) {
    const int IN_dim = 310;
    const int N  = in_sizes[0] / IN_dim;       // 100000
    const int E  = in_sizes[1] / 2;            // 1600000
    const int ET = E + N;                      // + self loops
    const int B  = out_size;                   // OUT == 1

    const float* x     = (const float*)d_in[0];
    const int*   ei    = (const int*)d_in[1];
    const int*   batch = (const int*)d_in[2];
    const float* W1  = (const float*)d_in[3];
    const float* as1 = (const float*)d_in[4];
    const float* ad1 = (const float*)d_in[5];
    const float* b1  = (const float*)d_in[6];
    const float* W2  = (const float*)d_in[7];
    const float* as2 = (const float*)d_in[8];
    const float* ad2 = (const float*)d_in[9];
    const float* b2  = (const float*)d_in[10];
    const float* Wn  = (const float*)d_in[11];
    const float* bn  = (const float*)d_in[12];
    const float* Wr  = (const float*)d_in[13];
    const float* br  = (const float*)d_in[14];
    const float* Wc  = (const float*)d_in[15];
    const float* bc  = (const float*)d_in[16];
    const int* srcp = ei;
    const int* dstp = ei + E;
    float* out = (float*)d_out;

    // ---- workspace carve-up (~60 MB) ----
    float* xp    = (float*)d_ws;               // [N*64]  projected features
    float* hbuf  = xp   + (size_t)N * HIDC;    // [N*64]  aggregated features
    float* als   = hbuf + (size_t)N * HIDC;    // [N]
    float* ald   = als  + N;                   // [N]
    float* denom = ald  + N;                   // [N]
    unsigned* emax = (unsigned*)(denom + N);   // [N]
    float* ew    = (float*)(emax + N);         // [ET]  e, then w, in place
    float* sums  = ew + ET;                    // [B*64]
    float* cnt   = sums + (size_t)B * HIDC;    // [B]
    int*   rootp = (int*)(cnt + B);            // [B]
    (void)ws_size; (void)n_in;

    const unsigned T = 256;
    const unsigned gemmBlocks = cdiv(cdiv(N, 16) * 4, T / 32);   // 8 waves/block
    const unsigned nB   = cdiv(N, T);
    const unsigned n64B = cdiv((unsigned)N * HIDC, T);
    const unsigned eB   = cdiv(ET, T);
    const unsigned e64B = cdiv((unsigned)ET * HIDC, T);

    // ================= GAT layer 1 =================
    gemm_wmma_f32<<<gemmBlocks, T, 0, stream>>>(x, W1, xp, N, IN_dim);
    node_attn<<<nB, T, 0, stream>>>(xp, as1, ad1, als, ald, N);
    set_u32<<<nB, T, 0, stream>>>(emax, INF_KEY, N);
    set_f32<<<nB, T, 0, stream>>>(denom, 0.0f, N);
    edge_max<<<eB, T, 0, stream>>>(srcp, dstp, als, ald, ew, emax, E, ET);
    edge_exp<<<eB, T, 0, stream>>>(srcp, dstp, ew, emax, denom, E, ET);
    recip<<<nB, T, 0, stream>>>(denom, N);
    set_f32<<<n64B, T, 0, stream>>>(hbuf, 0.0f, (unsigned)N * HIDC);
    edge_aggr<<<e64B, T, 0, stream>>>(srcp, dstp, ew, denom, xp, hbuf, E, ET);
    bias_relu<<<n64B, T, 0, stream>>>(hbuf, b1, (unsigned)N * HIDC);

    // ================= GAT layer 2 =================
    gemm_wmma_f32<<<gemmBlocks, T, 0, stream>>>(hbuf, W2, xp, N, HIDC);
    node_attn<<<nB, T, 0, stream>>>(xp, as2, ad2, als, ald, N);
    set_u32<<<nB, T, 0, stream>>>(emax, INF_KEY, N);
    set_f32<<<nB, T, 0, stream>>>(denom, 0.0f, N);
    edge_max<<<eB, T, 0, stream>>>(srcp, dstp, als, ald, ew, emax, E, ET);
    edge_exp<<<eB, T, 0, stream>>>(srcp, dstp, ew, emax, denom, E, ET);
    recip<<<nB, T, 0, stream>>>(denom, N);
    set_f32<<<n64B, T, 0, stream>>>(hbuf, 0.0f, (unsigned)N * HIDC);
    edge_aggr<<<e64B, T, 0, stream>>>(srcp, dstp, ew, denom, xp, hbuf, E, ET);
    bias_relu<<<n64B, T, 0, stream>>>(hbuf, b2, (unsigned)N * HIDC);

    // ================= readout =================
    set_f32<<<cdiv((unsigned)B * HIDC, T), T, 0, stream>>>(sums, 0.0f, (unsigned)B * HIDC);
    set_f32<<<cdiv((unsigned)B, T), T, 0, stream>>>(cnt, 0.0f, (unsigned)B);
    pool_accum<<<n64B, T, 0, stream>>>(hbuf, batch, sums, cnt, N);
    root_find<<<nB, T, 0, stream>>>(batch, rootp, N);
    readout<<<B, HIDC, 0, stream>>>(sums, cnt, x, rootp, Wr, br, Wn, bn, Wc, bc, out, IN_dim);
}